// DGCRNCell_7593502179664
// MI455X (gfx1250) — compile-verified
//
#include <hip/hip_runtime.h>
#include <hip/hip_bf16.h>

typedef float v2f __attribute__((ext_vector_type(2)));
typedef float v8f __attribute__((ext_vector_type(8)));

#define N_NODES 4096
#define NB      16      // batch
#define D_INF   32
#define D_H     128
#define F0      160     // D_INF + D_H
#define NC      2560    // NB * F0  (columns of the hop GEMM)
#define KDIM    480     // 3 * F0
#define LN_EPS  1e-5f

// D = A(16x4 f32) * B(4x16 f32) + C(16x16 f32), wave32
#define WMMA_F32(a, b, c) \
  __builtin_amdgcn_wmma_f32_16x16x4_f32(false, (a), false, (b), (short)0, (c), false, false)

__device__ __forceinline__ float sigmoidf_(float x) { return 1.0f / (1.0f + __expf(-x)); }

// 32-bit LDS byte offset of a generic pointer that points into LDS
__device__ __forceinline__ unsigned lds_offset(const void* p) {
  return (unsigned)(unsigned long long)(__attribute__((address_space(3))) const char*)p;
}

// ---------------------------------------------------------------------------
// Pack xh = concat(x_t, h_prev) into X[m, b*160+f]  (m-major for the hop GEMM)
// ---------------------------------------------------------------------------
__global__ __launch_bounds__(256) void pack_xh_kernel(
    const float* __restrict__ x_t, const float* __restrict__ h_prev,
    float* __restrict__ XH) {
  int idx = blockIdx.x * 256 + threadIdx.x;           // over N_NODES*NC
  int m   = idx / NC;
  int rem = idx - m * NC;
  int b   = rem / F0;
  int f   = rem - b * F0;
  float v = (f < D_INF) ? x_t[(b * N_NODES + m) * D_INF + f]
                        : h_prev[(b * N_NODES + m) * D_H + (f - D_INF)];
  XH[idx] = v;
}

// ---------------------------------------------------------------------------
// Hop GEMM: C[4096,2560] = adj[4096,4096] @ X[4096,2560]   (fp32 WMMA 16x16x4)
// 128x128 block tile, BK=16, 8 waves (4M x 2N), each wave owns 32x64.
// Double-buffered LDS; A tile staged via async-to-LDS (ASYNCcnt), B tile
// staged through VGPRs (transposed on the way in). One barrier per K-chunk.
// LDS row stride 20 floats: 16B-aligned rows for b128 async, conflict-free
// 8B frag reads.
// ---------------------------------------------------------------------------
#define AS_STRIDE 20
#define HALF_BUF  5120            // floats: As 128*20 + Bs 128*20
__global__ __launch_bounds__(256) void hop_gemm_kernel(
    const float* __restrict__ A, const float* __restrict__ X,
    float* __restrict__ C) {
  __shared__ __align__(16) float lds[2 * HALF_BUF];

  const int t     = threadIdx.x;
  const int lane  = t & 31;
  const int w     = t >> 5;
  const int lr    = lane & 15;           // N / M index within frag
  const int koff  = (lane >> 4) << 1;    // lanes 16-31 hold K+2,K+3
  const int mhalf = (lane >> 4) << 3;    // C rows: lanes 16-31 are M+8
  const int bm    = blockIdx.y * 128;
  const int bn    = blockIdx.x * 128;
  const int mW    = (w & 3) * 32;        // wave's M offset in tile
  const int nW    = (w >> 2) * 64;       // wave's N offset in tile

  // staging indices
  const int arow = t >> 2;               // 0..63
  const int ac4  = (t & 3) << 2;         // 0,4,8,12
  const int bk   = t >> 5;               // 0..7
  const int bc4  = (t & 31) << 2;        // 0..124

  v8f acc[2][4] = {};

  // ---- stage K-chunk `kb` into buffer half starting at float index `off`
  auto stageA = [&](int kb, int off) {
#pragma unroll
    for (int p = 0; p < 2; ++p) {
      const float* gp = &A[(bm + p * 64 + arow) * N_NODES + kb + ac4];
      unsigned lo = lds_offset(&lds[off + (p * 64 + arow) * AS_STRIDE + ac4]);
      asm volatile("global_load_async_to_lds_b128 %0, %1, off"
                   :: "v"(lo), "v"(gp) : "memory");
    }
  };

  int buf = 0;
  // ---- prologue: chunk 0
  stageA(0, 0);
  {
    float4 v0 = *(const float4*)&X[(bk + 0) * NC + bn + bc4];
    float4 v1 = *(const float4*)&X[(bk + 8) * NC + bn + bc4];
#pragma unroll
    for (int j = 0; j < 4; ++j) {
      lds[2560 + (bc4 + j) * AS_STRIDE + bk + 0] = ((const float*)&v0)[j];
      lds[2560 + (bc4 + j) * AS_STRIDE + bk + 8] = ((const float*)&v1)[j];
    }
  }
  asm volatile("s_wait_asynccnt 0" ::: "memory");
  __syncthreads();

  for (int kb = 16; kb < N_NODES; kb += 16) {
    const int cur = buf * HALF_BUF;
    const int nxt = (buf ^ 1) * HALF_BUF;

    // issue async A loads for next chunk (writes other buffer)
    stageA(kb, nxt);
    // global loads for next B chunk
    float4 v0 = *(const float4*)&X[(kb + bk + 0) * NC + bn + bc4];
    float4 v1 = *(const float4*)&X[(kb + bk + 8) * NC + bn + bc4];

    // compute current chunk
#pragma unroll
    for (int k0 = 0; k0 < 16; k0 += 4) {
      v2f a0 = *(const v2f*)&lds[cur + (mW + 0  + lr) * AS_STRIDE + k0 + koff];
      v2f a1 = *(const v2f*)&lds[cur + (mW + 16 + lr) * AS_STRIDE + k0 + koff];
#pragma unroll
      for (int j = 0; j < 4; ++j) {
        v2f b = *(const v2f*)&lds[cur + 2560 + (nW + j * 16 + lr) * AS_STRIDE + k0 + koff];
        acc[0][j] = WMMA_F32(a0, b, acc[0][j]);
        acc[1][j] = WMMA_F32(a1, b, acc[1][j]);
      }
    }

    // store next B chunk (transposed) into other buffer
#pragma unroll
    for (int j = 0; j < 4; ++j) {
      lds[nxt + 2560 + (bc4 + j) * AS_STRIDE + bk + 0] = ((const float*)&v0)[j];
      lds[nxt + 2560 + (bc4 + j) * AS_STRIDE + bk + 8] = ((const float*)&v1)[j];
    }

    asm volatile("s_wait_asynccnt 0" ::: "memory");
    __syncthreads();
    buf ^= 1;
  }

  // ---- last chunk compute
  {
    const int cur = buf * HALF_BUF;
#pragma unroll
    for (int k0 = 0; k0 < 16; k0 += 4) {
      v2f a0 = *(const v2f*)&lds[cur + (mW + 0  + lr) * AS_STRIDE + k0 + koff];
      v2f a1 = *(const v2f*)&lds[cur + (mW + 16 + lr) * AS_STRIDE + k0 + koff];
#pragma unroll
      for (int j = 0; j < 4; ++j) {
        v2f b = *(const v2f*)&lds[cur + 2560 + (nW + j * 16 + lr) * AS_STRIDE + k0 + koff];
        acc[0][j] = WMMA_F32(a0, b, acc[0][j]);
        acc[1][j] = WMMA_F32(a1, b, acc[1][j]);
      }
    }
  }

#pragma unroll
  for (int i = 0; i < 2; ++i)
#pragma unroll
    for (int j = 0; j < 4; ++j) {
      int col = bn + nW + j * 16 + lr;
#pragma unroll
      for (int r = 0; r < 8; ++r) {
        int row = bm + mW + i * 16 + mhalf + r;
        C[row * NC + col] = acc[i][j][r];
      }
    }
}

// ---------------------------------------------------------------------------
// z/r GEMM: one block per node n. M-tile rows = the 16 batches (A row for
// (n,b) is XH/H1/H2[n, b*160 + f] -> staged whole-row in LDS).
// Fused: z = sigmoid(..)->Z ; r = sigmoid(..) and cand_xh h-part written
// in-place into XH (x-part already holds x_t, shared with cand_xh).
// ---------------------------------------------------------------------------
__global__ __launch_bounds__(256) void zr_kernel(
    const float* __restrict__ XH, const float* __restrict__ H1,
    const float* __restrict__ H2,
    const float* __restrict__ Wz, const float* __restrict__ bz,
    const float* __restrict__ Wr, const float* __restrict__ br,
    const float* __restrict__ h_prev,
    float* __restrict__ Z, float* __restrict__ CXH) {
  __shared__ __align__(16) float g[3 * NC];   // 30 KB: row n of XH,H1,H2
  const int n = blockIdx.x;
  const int t = threadIdx.x;
#pragma unroll
  for (int p = 0; p < 10; ++p) {
    g[p * 256 + t]          = XH[n * NC + p * 256 + t];
    g[NC + p * 256 + t]     = H1[n * NC + p * 256 + t];
    g[2 * NC + p * 256 + t] = H2[n * NC + p * 256 + t];
  }
  __syncthreads();

  const int lane  = t & 31;
  const int w     = t >> 5;
  const int lr    = lane & 15;
  const int koff  = (lane >> 4) << 1;
  const int mhalf = (lane >> 4) << 3;
  const int c     = w * 16 + lr;          // output feature column 0..127
  const float* wzRow = &Wz[c * KDIM];     // B[k][c] = W[c][k]: contiguous in k
  const float* wrRow = &Wr[c * KDIM];

  v8f accZ = {}, accR = {};
  for (int kk = 0; kk < KDIM; kk += 4) {
    int seg = kk / F0;                    // which of XH/H1/H2 (160 % 4 == 0)
    int f0  = kk - seg * F0;
    v2f a   = *(const v2f*)&g[seg * NC + lr * F0 + f0 + koff];
    v2f bzf = *(const v2f*)&wzRow[kk + koff];
    v2f brf = *(const v2f*)&wrRow[kk + koff];
    accZ = WMMA_F32(a, bzf, accZ);
    accR = WMMA_F32(a, brf, accR);
  }

  const float bzc = bz[c], brc = br[c];
#pragma unroll
  for (int r = 0; r < 8; ++r) {
    int b = mhalf + r;
    float zv = sigmoidf_(accZ[r] + bzc);
    Z[(b * N_NODES + n) * D_H + c] = zv;
    float rv = sigmoidf_(accR[r] + brc);
    CXH[n * NC + b * F0 + D_INF + c] = rv * h_prev[(b * N_NODES + n) * D_H + c];
  }
}

// ---------------------------------------------------------------------------
// Final: h_tilde GEMM + tanh + gate combine + per-row LayerNorm -> out
// ---------------------------------------------------------------------------
__global__ __launch_bounds__(256) void final_kernel(
    const float* __restrict__ CXH, const float* __restrict__ H1,
    const float* __restrict__ H2,
    const float* __restrict__ Wh, const float* __restrict__ bh,
    const float* __restrict__ Z, const float* __restrict__ h_prev,
    const float* __restrict__ gamma, const float* __restrict__ beta,
    float* __restrict__ out) {
  __shared__ __align__(16) float g[3 * NC];   // reused later as h tile + stats
  const int n = blockIdx.x;
  const int t = threadIdx.x;
#pragma unroll
  for (int p = 0; p < 10; ++p) {
    g[p * 256 + t]          = CXH[n * NC + p * 256 + t];
    g[NC + p * 256 + t]     = H1[n * NC + p * 256 + t];
    g[2 * NC + p * 256 + t] = H2[n * NC + p * 256 + t];
  }
  __syncthreads();

  const int lane  = t & 31;
  const int w     = t >> 5;
  const int lr    = lane & 15;
  const int koff  = (lane >> 4) << 1;
  const int mhalf = (lane >> 4) << 3;
  const int c     = w * 16 + lr;
  const float* whRow = &Wh[c * KDIM];

  v8f acc = {};
  for (int kk = 0; kk < KDIM; kk += 4) {
    int seg = kk / F0;
    int f0  = kk - seg * F0;
    v2f a   = *(const v2f*)&g[seg * NC + lr * F0 + f0 + koff];
    v2f bf  = *(const v2f*)&whRow[kk + koff];
    acc = WMMA_F32(a, bf, acc);
  }
  __syncthreads();   // done reading g; reuse as h tile [16][128]

  const float bhc = bh[c];
#pragma unroll
  for (int r = 0; r < 8; ++r) {
    int b = mhalf + r;
    float ht = tanhf(acc[r] + bhc);
    float hp = h_prev[(b * N_NODES + n) * D_H + c];
    float zv = Z[(b * N_NODES + n) * D_H + c];
    g[b * D_H + c] = (1.0f - zv) * hp + zv * ht;
  }
  __syncthreads();

  if (t < 16) {                // per-batch LayerNorm stats over 128 features
    float mu = 0.0f;
#pragma unroll 8
    for (int j = 0; j < D_H; ++j) mu += g[t * D_H + j];
    mu *= (1.0f / D_H);
    float var = 0.0f;
#pragma unroll 8
    for (int j = 0; j < D_H; ++j) { float d = g[t * D_H + j] - mu; var += d * d; }
    var *= (1.0f / D_H);
    g[2048 + t] = mu;
    g[2080 + t] = rsqrtf(var + LN_EPS);
  }
  __syncthreads();

#pragma unroll
  for (int q = 0; q < 8; ++q) {
    int e = q * 256 + t;       // 2048 elements
    int b = e >> 7, cc = e & 127;
    out[(b * N_NODES + n) * D_H + cc] =
        (g[e] - g[2048 + b]) * g[2080 + b] * gamma[cc] + beta[cc];
  }
}

// ---------------------------------------------------------------------------
extern "C" void kernel_launch(void* const* d_in, const int* in_sizes, int n_in,
                              void* d_out, int out_size, void* d_ws, size_t ws_size,
                              hipStream_t stream) {
  const float* x_t    = (const float*)d_in[0];
  const float* h_prev = (const float*)d_in[1];
  const float* adj    = (const float*)d_in[2];
  const float* Wz     = (const float*)d_in[3];
  const float* bz     = (const float*)d_in[4];
  const float* Wr     = (const float*)d_in[5];
  const float* br     = (const float*)d_in[6];
  const float* Wh     = (const float*)d_in[7];
  const float* bh     = (const float*)d_in[8];
  const float* gamma  = (const float*)d_in[9];
  const float* beta   = (const float*)d_in[10];
  float* out = (float*)d_out;

  float* XH = (float*)d_ws;                       // [4096, 2560]  40 MB
  float* H1 = XH + (size_t)N_NODES * NC;          // [4096, 2560]  40 MB
  float* H2 = H1 + (size_t)N_NODES * NC;          // [4096, 2560]  40 MB
  float* Z  = H2 + (size_t)N_NODES * NC;          // [16,4096,128] 32 MB

  dim3 hopGrid(NC / 128, N_NODES / 128);          // 20 x 32

  pack_xh_kernel<<<(N_NODES * NC) / 256, 256, 0, stream>>>(x_t, h_prev, XH);
  hop_gemm_kernel<<<hopGrid, 256, 0, stream>>>(adj, XH, H1);   // A @ xh
  hop_gemm_kernel<<<hopGrid, 256, 0, stream>>>(adj, H1, H2);   // A^2 @ xh
  zr_kernel<<<N_NODES, 256, 0, stream>>>(XH, H1, H2, Wz, bz, Wr, br, h_prev,
                                         Z, XH /* cand_xh in place */);
  hop_gemm_kernel<<<hopGrid, 256, 0, stream>>>(adj, XH, H1);   // A @ cand
  hop_gemm_kernel<<<hopGrid, 256, 0, stream>>>(adj, H1, H2);   // A^2 @ cand
  final_kernel<<<N_NODES, 256, 0, stream>>>(XH, H1, H2, Wh, bh, Z, h_prev,
                                            gamma, beta, out);
}